// CvxSolver_32916629356744
// MI455X (gfx1250) — compile-verified
//
#include <hip/hip_runtime.h>

typedef __attribute__((ext_vector_type(2))) float        v2f;
typedef __attribute__((ext_vector_type(8))) float        v8f;
typedef __attribute__((ext_vector_type(4))) unsigned int v4u;
typedef __attribute__((ext_vector_type(4))) int          v4i;
typedef __attribute__((ext_vector_type(8))) int          v8i;

#define BATCH   128
#define MDIM    128
#define NDIM    512
#define ITERS   40
#define SIGMA_C 0.1f
#define TAU_C   0.995f
#define KC      32
#define NCHUNK  (NDIM / KC)          // 16
#define GS_STRIDE 33                 // 32 dwords + 1 pad dword per row
#define MS_STRIDE 129

// float offsets inside dynamic LDS
#define OFF_MS   0
#define OFF_GS0  (MDIM * MS_STRIDE)                  // 16512
#define OFF_GS1  (OFF_GS0 + MDIM * GS_STRIDE)        // +4224
#define OFF_VEC  (OFF_GS1 + MDIM * GS_STRIDE)        // +4224
// 7 n-vectors + 3 m-vectors + red + scal
#define SM_FLOATS (OFF_VEC + 7 * NDIM + 3 * MDIM + 256 + 4)

#if __has_builtin(__builtin_amdgcn_tensor_load_to_lds)
#define HAVE_TDM 1
#else
#define HAVE_TDM 0
#endif

#if HAVE_TDM
// Issue one TDM 2-D tile load: 32x128 dwords of A (row stride 512 dwords)
// into LDS at byte offset lds_byte, with 1 pad dword inserted after every
// 32 dwords (-> LDS row stride 33 dwords, matches GS_STRIDE).
__device__ __forceinline__ void tdm_load_chunk(const float* gptr, unsigned lds_byte) {
    unsigned long long ga = (unsigned long long)(uintptr_t)gptr;
    // D# group0: count=1 | lds_addr[63:32] | global_addr[120:64] | type=2 [127:126]
    v4u g0 = { 1u,
               lds_byte,
               (unsigned)(ga & 0xffffffffu),
               (unsigned)((ga >> 32) & 0x01ffffffu) | (2u << 30) };
    // D# group1
    unsigned w0 = (2u << 16)      // data_size = 4 bytes
                | (1u << 20)      // pad_enable
                | (4u << 22)      // pad_interval: 32 dwords
                | (0u << 25);     // pad_amount: 1 dword
    v8i g1 = { (int)w0,
               (int)((512u & 0xffffu) << 16),                    // tensor_dim0 lo16 -> bits[79:64... w1[31:16]]
               (int)(((512u >> 16) & 0xffffu) | ((128u & 0xffffu) << 16)), // dim0 hi16 | tensor_dim1 lo16
               (int)(((128u >> 16) & 0xffffu) | (32u << 16)),    // dim1 hi16 | tile_dim0 = 32
               (int)(128u),                                      // tile_dim1 = 128, tile_dim2 = 0
               (int)(512u),                                      // tensor_dim0_stride lo32 = 512
               0, 0 };                                           // stride0 hi / dim1_stride unused (2-D)
    v4i z4 = {0, 0, 0, 0};
#if __clang_major__ >= 23
    v8i z8 = {0, 0, 0, 0, 0, 0, 0, 0};
    __builtin_amdgcn_tensor_load_to_lds(g0, g1, z4, z4, z8, 0);
#else
    __builtin_amdgcn_tensor_load_to_lds(g0, g1, z4, z4, 0);
#endif
}
#endif

__launch_bounds__(256, 1)
__global__ void ipm_kernel(const float* __restrict__ A,
                           const float* __restrict__ bvec,
                           const float* __restrict__ cvec,
                           float* __restrict__ out)
{
    extern __shared__ float sm[];
    float* Ms   = sm + OFF_MS;       // 128*129 Cholesky workspace
    float* Gs0  = sm + OFF_GS0;      // 128*33  A chunk buffer 0 (raw A)
    float* Gs1  = sm + OFF_GS1;      // 128*33  A chunk buffer 1 (raw A)
    float* xv   = sm + OFF_VEC;      // 512
    float* sv   = xv   + NDIM;       // 512
    float* dvec = sv   + NDIM;       // 512  d = x/s
    float* rdv  = dvec + NDIM;       // 512  dual residual
    float* rhsx = rdv  + NDIM;       // 512
    float* dxv  = rhsx + NDIM;       // 512
    float* dsvv = dxv  + NDIM;       // 512
    float* yv   = dsvv + NDIM;       // 128
    float* rpv  = yv   + MDIM;       // 128  primal residual
    float* zv   = rpv  + MDIM;       // 128  rhs_y / solve vec / dy
    float* red  = zv   + MDIM;       // 256  reduction scratch
    float* scal = red  + 256;        // 4    broadcast scalars

    const int tid  = threadIdx.x;
    const int lane = tid & 31;
    const int wv   = tid >> 5;       // wave id 0..7
    const int half = lane >> 4;
    const int lr   = lane & 15;
    const int bid  = blockIdx.x;

    const float* Ab = A    + (size_t)bid * MDIM * NDIM;
    const float* bb = bvec + (size_t)bid * MDIM;
    const float* cb = cvec + (size_t)bid * NDIM;

#if HAVE_TDM
    const unsigned lds_base = (unsigned)__builtin_amdgcn_groupstaticsize();
    const unsigned gs_byte[2] = { lds_base + OFF_GS0 * 4u, lds_base + OFF_GS1 * 4u };
#endif

    for (int n = tid; n < NDIM; n += 256) { xv[n] = 1.0f; sv[n] = 1.0f; }
    for (int m = tid; m < MDIM; m += 256) yv[m] = 0.0f;
    __syncthreads();

#pragma unroll 1
    for (int it = 0; it < ITERS; ++it) {
#if HAVE_TDM
        // Kick off DMA of chunk 0 early: overlaps with residual/mu/d work.
        if (wv == 0) tdm_load_chunk(Ab + 0, gs_byte[0]);
#endif
        // ---- r_p = A x - b  (wave handles 16 rows, lane-parallel dot) ----
        for (int r = wv * 16; r < wv * 16 + 16; ++r) {
            float acc = 0.f;
            for (int n = lane; n < NDIM; n += 32) acc += Ab[r * NDIM + n] * xv[n];
            for (int off = 16; off > 0; off >>= 1) acc += __shfl_down(acc, off, 32);
            if (lane == 0) rpv[r] = acc - bb[r];
        }
        // ---- r_d = A^T y + s - c  (column per thread, coalesced over rows) ----
        for (int n = tid; n < NDIM; n += 256) {
            float acc = 0.f;
            for (int m = 0; m < MDIM; ++m) acc += Ab[m * NDIM + n] * yv[m];
            rdv[n] = acc + sv[n] - cb[n];
        }
        __syncthreads();

        // ---- mu = mean(x*s) ----
        {
            float p = 0.f;
            for (int n = tid; n < NDIM; n += 256) p += xv[n] * sv[n];
            red[tid] = p; __syncthreads();
            for (int off = 128; off > 0; off >>= 1) {
                if (tid < off) red[tid] += red[tid + off];
                __syncthreads();
            }
            if (tid == 0) scal[0] = red[0] * (1.0f / (float)NDIM);
            __syncthreads();
        }
        const float mu = scal[0];

        // ---- d, rhs_x ----
        for (int n = tid; n < NDIM; n += 256) {
            float d = xv[n] / sv[n];
            dvec[n] = d;
            rhsx[n] = d * rdv[n] - xv[n] + SIGMA_C * mu / sv[n];
        }
        __syncthreads();

        // ---- M = A D A^T via V_WMMA_F32_16X16X4_F32 ----
        // a-fragment carries A*d, b-fragment carries raw A.
        // wave wv owns output row-block wv: tiles (wv, t), t=0..7.
        v8f acc[8] = {};
#pragma unroll 1
        for (int c = 0; c < NCHUNK; ++c) {
            const int k0 = c * KC;
#if HAVE_TDM
            if (wv == 0) {
                if (c + 1 < NCHUNK) {
                    tdm_load_chunk(Ab + (c + 1) * KC, gs_byte[(c + 1) & 1]);
                    __builtin_amdgcn_s_wait_tensorcnt(1);   // chunk c landed
                } else {
                    __builtin_amdgcn_s_wait_tensorcnt(0);
                }
            }
            __syncthreads();
            const float* G = (c & 1) ? Gs1 : Gs0;
#else
            // Fallback: synchronous cooperative staging into buffer 0.
            for (int idx = tid; idx < MDIM * KC; idx += 256) {
                int row = idx / KC;
                int col = idx % KC;
                Gs0[row * GS_STRIDE + col] = Ab[row * NDIM + k0 + col];
            }
            __syncthreads();
            const float* G = Gs0;
#endif
            const int arow = wv * 16 + lr;
            for (int kk = 0; kk < KC; kk += 4) {
                const int kg = k0 + kk + 2 * half;
                v2f af;
                af.x = G[arow * GS_STRIDE + kk + 2 * half]     * dvec[kg];
                af.y = G[arow * GS_STRIDE + kk + 2 * half + 1] * dvec[kg + 1];
#pragma unroll
                for (int t = 0; t < 8; ++t) {
                    const int brow = t * 16 + lr;
                    v2f bf;
                    bf.x = G[brow * GS_STRIDE + kk + 2 * half];
                    bf.y = G[brow * GS_STRIDE + kk + 2 * half + 1];
                    acc[t] = __builtin_amdgcn_wmma_f32_16x16x4_f32(
                        false, af, false, bf, (short)0, acc[t], false, false);
                }
            }
            __syncthreads();   // buffer c&1 free for chunk c+2 overwrite
        }
        // write M tiles to LDS (+1e-8 on the diagonal)
#pragma unroll
        for (int t = 0; t < 8; ++t) {
#pragma unroll
            for (int v = 0; v < 8; ++v) {
                int row = wv * 16 + v + 8 * half;   // C/D layout: vgpr v, lane half
                int col = t * 16 + lr;
                float val = acc[t][v];
                if (row == col) val += 1e-8f;
                Ms[row * MS_STRIDE + col] = val;
            }
        }
        // ---- rhs_y = -r_p - A rhs_x ----
        for (int r = wv * 16; r < wv * 16 + 16; ++r) {
            float a2 = 0.f;
            for (int n = lane; n < NDIM; n += 32) a2 += Ab[r * NDIM + n] * rhsx[n];
            for (int off = 16; off > 0; off >>= 1) a2 += __shfl_down(a2, off, 32);
            if (lane == 0) zv[r] = -rpv[r] - a2;
        }
        __syncthreads();

        // ---- Cholesky (lower, in place) ----
        for (int k = 0; k < MDIM; ++k) {
            if (tid == 0) Ms[k * MS_STRIDE + k] = sqrtf(Ms[k * MS_STRIDE + k]);
            __syncthreads();
            const float inv = 1.0f / Ms[k * MS_STRIDE + k];
            for (int i = k + 1 + tid; i < MDIM; i += 256) Ms[i * MS_STRIDE + k] *= inv;
            __syncthreads();
            for (int j = k + 1; j < MDIM; ++j) {
                const float ljk = Ms[j * MS_STRIDE + k];
                for (int i = j + tid; i < MDIM; i += 256)
                    Ms[i * MS_STRIDE + j] -= Ms[i * MS_STRIDE + k] * ljk;
            }
            __syncthreads();
        }
        // ---- forward solve L z = rhs_y (column-oriented) ----
        for (int k = 0; k < MDIM; ++k) {
            if (tid == 0) zv[k] = zv[k] / Ms[k * MS_STRIDE + k];
            __syncthreads();
            const float zk = zv[k];
            for (int i = k + 1 + tid; i < MDIM; i += 256)
                zv[i] -= Ms[i * MS_STRIDE + k] * zk;
            __syncthreads();
        }
        // ---- backward solve L^T dy = z ----
        for (int k = MDIM - 1; k >= 0; --k) {
            if (tid == 0) zv[k] = zv[k] / Ms[k * MS_STRIDE + k];
            __syncthreads();
            const float zk = zv[k];
            for (int i = tid; i < k; i += 256)
                zv[i] -= Ms[k * MS_STRIDE + i] * zk;    // L^T[i][k] = L[k][i]
            __syncthreads();
        }
        // zv now holds dy

        // ---- Aty = A^T dy (consumed locally); dx ; ds ----
        for (int n = tid; n < NDIM; n += 256) {
            float aty = 0.f;
            for (int m = 0; m < MDIM; ++m) aty += Ab[m * NDIM + n] * zv[m];
            dxv[n]  = dvec[n] * aty + rhsx[n];
            dsvv[n] = -rdv[n] - aty;
        }
        __syncthreads();

        // ---- fraction-to-the-boundary: ap ----
        {
            float mn = 1e30f;
            for (int n = tid; n < NDIM; n += 256)
                if (dxv[n] < 0.f) mn = fminf(mn, -xv[n] / dxv[n]);
            red[tid] = mn; __syncthreads();
            for (int off = 128; off > 0; off >>= 1) {
                if (tid < off) red[tid] = fminf(red[tid], red[tid + off]);
                __syncthreads();
            }
            if (tid == 0) scal[1] = fminf(1.0f, TAU_C * red[0]);
            __syncthreads();
        }
        // ---- ad ----
        {
            float mn = 1e30f;
            for (int n = tid; n < NDIM; n += 256)
                if (dsvv[n] < 0.f) mn = fminf(mn, -sv[n] / dsvv[n]);
            red[tid] = mn; __syncthreads();
            for (int off = 128; off > 0; off >>= 1) {
                if (tid < off) red[tid] = fminf(red[tid], red[tid + off]);
                __syncthreads();
            }
            if (tid == 0) scal[2] = fminf(1.0f, TAU_C * red[0]);
            __syncthreads();
        }
        const float ap = scal[1];
        const float ad = scal[2];
        for (int n = tid; n < NDIM; n += 256) {
            xv[n] += ap * dxv[n];
            sv[n] += ad * dsvv[n];
        }
        for (int m = tid; m < MDIM; m += 256) yv[m] += ad * zv[m];
        __syncthreads();
    }

    for (int n = tid; n < NDIM; n += 256)
        out[(size_t)bid * NDIM + n] = xv[n];
}

extern "C" void kernel_launch(void* const* d_in, const int* in_sizes, int n_in,
                              void* d_out, int out_size, void* d_ws, size_t ws_size,
                              hipStream_t stream) {
    (void)in_sizes; (void)n_in; (void)out_size; (void)d_ws; (void)ws_size;
    const float* A = (const float*)d_in[0];
    const float* b = (const float*)d_in[1];
    const float* c = (const float*)d_in[2];
    float* out = (float*)d_out;
    const size_t shmem = (size_t)SM_FLOATS * sizeof(float);
    ipm_kernel<<<dim3(BATCH), dim3(256), shmem, stream>>>(A, b, c, out);
}